// GAT_20512763806293
// MI455X (gfx1250) — compile-verified
//
#include <hip/hip_runtime.h>
#include <math.h>

typedef __attribute__((ext_vector_type(2))) float v2f;
typedef __attribute__((ext_vector_type(8))) float v8f;

#define NEG_SLOPE 0.2f

// ---------- monotone float <-> uint mapping for atomicMax on floats ----------
__device__ __forceinline__ unsigned fmap(float f) {
    unsigned b = __float_as_uint(f);
    return (b & 0x80000000u) ? ~b : (b | 0x80000000u);
}
__device__ __forceinline__ float funmap(unsigned u) {
    unsigned b = (u & 0x80000000u) ? (u & 0x7FFFFFFFu) : ~u;
    return __uint_as_float(b);
}

// ---------------- WMMA f32 GEMM: C[nrows,NCOLS] = op(A)[nrows,K] @ B[K,NCOLS] ------
// One 16x16 output tile per wave; K stepped by 4 via V_WMMA_F32_16X16X4_F32.
// K/NCOLS are compile-time so all strided loads fold into immediate offsets.
template<int RELU_A, int K, int NCOLS>
__global__ void gemm_wmma_f32(const float* __restrict__ A, const float* __restrict__ B,
                              float* __restrict__ C, int nrows)
{
    constexpr int TILES_N = NCOLS / 16;
    int wave = (int)((blockIdx.x * blockDim.x + threadIdx.x) >> 5);
    int lane = threadIdx.x & 31;
    int tiles_m = (nrows + 15) >> 4;
    if (wave >= tiles_m * TILES_N) return;           // whole-wave uniform exit
    int tm = wave / TILES_N;                         // TILES_N is 16 or 4 -> shift
    int tn = wave - tm * TILES_N;

    int row  = (tm << 4) + (lane & 15);
    int rowc = row < nrows ? row : nrows - 1;        // clamp loads for tail tile
    int col  = (tn << 4) + (lane & 15);
    int kh   = (lane >> 4) << 1;                     // lanes 0-15: K {0,1}; 16-31: K {2,3}

    const float* ap = A + (size_t)rowc * K + kh;
    const float* bp = B + (size_t)kh * NCOLS + col;

    v8f acc = {0.f,0.f,0.f,0.f,0.f,0.f,0.f,0.f};
    #pragma unroll 16
    for (int k = 0; k < K; k += 4) {
        float a0 = ap[k], a1 = ap[k + 1];
        if (RELU_A) { a0 = fmaxf(a0, 0.f); a1 = fmaxf(a1, 0.f); }
        v2f a; a.x = a0; a.y = a1;
        v2f b; b.x = bp[k * NCOLS]; b.y = bp[(k + 1) * NCOLS];
        acc = __builtin_amdgcn_wmma_f32_16x16x4_f32(false, a, false, b,
                                                    (short)0, acc, false, false);
    }
    // C/D layout: VGPR r -> row r (lanes 0-15) / row r+8 (lanes 16-31); col = lane&15
    int rbase = (tm << 4) + ((lane >> 4) << 3);
    #pragma unroll
    for (int r = 0; r < 8; ++r) {
        int rr = rbase + r;
        if (rr < nrows) C[(size_t)rr * NCOLS + col] = acc[r];
    }
}

// ------------- per-(node,head) attention dot products: wave per (n,h) -------------
template<int H>
__global__ void att_dots(const float* __restrict__ Hm, const float* __restrict__ att_s,
                         const float* __restrict__ att_d, float* __restrict__ as_,
                         float* __restrict__ ad_, int nN)
{
    int w = (int)((blockIdx.x * blockDim.x + threadIdx.x) >> 5);
    int lane = threadIdx.x & 31;
    if (w >= nN * H) return;
    int n = w / H;
    int h = w - n * H;
    const float* hp = Hm + (size_t)n * (H * 64) + h * 64;
    float v0 = hp[lane], v1 = hp[lane + 32];
    float s = v0 * att_s[h * 64 + lane] + v1 * att_s[h * 64 + lane + 32];
    float d = v0 * att_d[h * 64 + lane] + v1 * att_d[h * 64 + lane + 32];
    #pragma unroll
    for (int off = 16; off; off >>= 1) {
        s += __shfl_xor(s, off, 32);
        d += __shfl_xor(d, off, 32);
    }
    if (lane == 0) { as_[w] = s; ad_[w] = d; }
}

// ------------- init: accumulator = bias broadcast, max = -inf(mapped 0), sum = 0 ---
template<int COLS, int H>
__global__ void init_layer(float* __restrict__ acc, const float* __restrict__ bias,
                           unsigned* __restrict__ mx, float* __restrict__ sm, int nN)
{
    size_t idx = (size_t)blockIdx.x * blockDim.x + threadIdx.x;
    size_t tot = (size_t)nN * COLS;
    if (idx < tot) acc[idx] = bias[idx & (COLS - 1)];
    if (idx < (size_t)nN * H) { mx[idx] = 0u; sm[idx] = 0.f; }
}

// ------------- edge pass 1: segment max of leaky-relu scores -----------------------
template<int H>
__global__ void edge_max(const int* __restrict__ src, const int* __restrict__ dst,
                         const float* __restrict__ as_, const float* __restrict__ ad_,
                         unsigned* __restrict__ mx, int nE)
{
    long long idx = (long long)blockIdx.x * blockDim.x + threadIdx.x;
    if (idx >= (long long)nE * H) return;
    int e = (int)(idx / H);
    int h = (int)(idx & (H - 1));
    int s = src[e], d = dst[e];
    float v = as_[s * H + h] + ad_[d * H + h];
    v = v >= 0.f ? v : NEG_SLOPE * v;
    atomicMax(&mx[d * H + h], fmap(v));
}

// ------------- edge pass 2: exp(e - max[dst]) + segment sum ------------------------
template<int H>
__global__ void edge_expsum(const int* __restrict__ src, const int* __restrict__ dst,
                            const float* __restrict__ as_, const float* __restrict__ ad_,
                            const unsigned* __restrict__ mx, float* __restrict__ sm,
                            float* __restrict__ eexp, int nE)
{
    long long idx = (long long)blockIdx.x * blockDim.x + threadIdx.x;
    if (idx >= (long long)nE * H) return;
    int e = (int)(idx / H);
    int h = (int)(idx & (H - 1));
    int s = src[e], d = dst[e];
    float v = as_[s * H + h] + ad_[d * H + h];
    v = v >= 0.f ? v : NEG_SLOPE * v;
    float ex = __expf(v - funmap(mx[d * H + h]));
    eexp[idx] = ex;
    atomicAdd(&sm[d * H + h], ex);
}

// ------------- edge pass 3: out[dst] += h[src] * alpha (one thread per (e,hc)) -----
template<int H, int C>   // C = H*64
__global__ void edge_msg(const int* __restrict__ src, const int* __restrict__ dst,
                         const float* __restrict__ eexp, const float* __restrict__ sm,
                         const float* __restrict__ Hm, float* __restrict__ acc, int nE)
{
    long long idx = (long long)blockIdx.x * blockDim.x + threadIdx.x;
    if (idx >= (long long)nE * C) return;
    int e  = (int)(idx / C);
    int hc = (int)(idx & (C - 1));
    int h  = hc >> 6;
    int s = src[e], d = dst[e];
    float alpha = eexp[(size_t)e * H + h] / sm[(size_t)d * H + h];
    atomicAdd(&acc[(size_t)d * C + hc], Hm[(size_t)s * C + hc] * alpha);
}

// -----------------------------------------------------------------------------------
extern "C" void kernel_launch(void* const* d_in, const int* in_sizes, int n_in,
                              void* d_out, int out_size, void* d_ws, size_t ws_size,
                              hipStream_t stream)
{
    const float* x        = (const float*)d_in[0];
    const int*   ei       = (const int*)  d_in[1];
    const float* W1       = (const float*)d_in[2];
    const float* att_src1 = (const float*)d_in[3];
    const float* att_dst1 = (const float*)d_in[4];
    const float* b1       = (const float*)d_in[5];
    const float* W2       = (const float*)d_in[6];
    const float* att_src2 = (const float*)d_in[7];
    const float* att_dst2 = (const float*)d_in[8];
    const float* b2       = (const float*)d_in[9];

    const int nN = in_sizes[0] / 64;
    const int nE = in_sizes[1] / 2;
    const int* src = ei;
    const int* dst = ei + nE;

    // ---------------- workspace carve-out (all L2-resident: ~135 MB) ----------------
    size_t off = 0;
    auto carve = [&](size_t bytes) -> void* {
        void* p = (char*)d_ws + off;
        off += (bytes + 255) & ~(size_t)255;
        return p;
    };
    float*    h1    = (float*)   carve((size_t)nN * 256 * 4);
    float*    out1  = (float*)   carve((size_t)nN * 256 * 4);
    float*    as1   = (float*)   carve((size_t)nN * 4 * 4);
    float*    ad1   = (float*)   carve((size_t)nN * 4 * 4);
    unsigned* mx1   = (unsigned*)carve((size_t)nN * 4 * 4);
    float*    sm1   = (float*)   carve((size_t)nN * 4 * 4);
    float*    eexp1 = (float*)   carve((size_t)nE * 4 * 4);
    float*    h2    = (float*)   carve((size_t)nN * 64 * 4);
    float*    as2   = (float*)   carve((size_t)nN * 4);
    float*    ad2   = (float*)   carve((size_t)nN * 4);
    unsigned* mx2   = (unsigned*)carve((size_t)nN * 4);
    float*    sm2   = (float*)   carve((size_t)nN * 4);
    float*    eexp2 = (float*)   carve((size_t)nE * 4);
    (void)ws_size; (void)n_in; (void)out_size;

    const int BLK = 256;
    auto blocks_for = [](long long work, int blk) -> unsigned {
        return (unsigned)((work + blk - 1) / blk);
    };

    // =============================== layer 1 (heads=4) ===============================
    {   // GEMM1: h1[N,256] = x[N,64] @ W1[64,256]
        long long waves = (long long)((nN + 15) / 16) * (256 / 16);
        gemm_wmma_f32<0, 64, 256><<<blocks_for(waves * 32, BLK), BLK, 0, stream>>>(x, W1, h1, nN);
    }
    att_dots<4><<<blocks_for((long long)nN * 4 * 32, BLK), BLK, 0, stream>>>(
        h1, att_src1, att_dst1, as1, ad1, nN);
    init_layer<256, 4><<<blocks_for((long long)nN * 256, BLK), BLK, 0, stream>>>(
        out1, b1, mx1, sm1, nN);
    edge_max<4><<<blocks_for((long long)nE * 4, BLK), BLK, 0, stream>>>(
        src, dst, as1, ad1, mx1, nE);
    edge_expsum<4><<<blocks_for((long long)nE * 4, BLK), BLK, 0, stream>>>(
        src, dst, as1, ad1, mx1, sm1, eexp1, nE);
    edge_msg<4, 256><<<blocks_for((long long)nE * 256, BLK), BLK, 0, stream>>>(
        src, dst, eexp1, sm1, h1, out1, nE);

    // =============================== layer 2 (heads=1) ===============================
    {   // GEMM2: h2[N,64] = relu(out1)[N,256] @ W2[256,64]   (ReLU fused on A load)
        long long waves = (long long)((nN + 15) / 16) * (64 / 16);
        gemm_wmma_f32<1, 256, 64><<<blocks_for(waves * 32, BLK), BLK, 0, stream>>>(out1, W2, h2, nN);
    }
    att_dots<1><<<blocks_for((long long)nN * 32, BLK), BLK, 0, stream>>>(
        h2, att_src2, att_dst2, as2, ad2, nN);
    init_layer<64, 1><<<blocks_for((long long)nN * 64, BLK), BLK, 0, stream>>>(
        (float*)d_out, b2, mx2, sm2, nN);
    edge_max<1><<<blocks_for((long long)nE, BLK), BLK, 0, stream>>>(
        src, dst, as2, ad2, mx2, nE);
    edge_expsum<1><<<blocks_for((long long)nE, BLK), BLK, 0, stream>>>(
        src, dst, as2, ad2, mx2, sm2, eexp2, nE);
    edge_msg<1, 64><<<blocks_for((long long)nE * 64, BLK), BLK, 0, stream>>>(
        src, dst, eexp2, sm2, h2, (float*)d_out, nE);
}